// BIGCF_49228915146895
// MI455X (gfx1250) — compile-verified
//
#include <hip/hip_runtime.h>

typedef __attribute__((ext_vector_type(2))) float v2f;
typedef __attribute__((ext_vector_type(8))) float v8f;

#define N_USERS   100000
#define N_ITEMS   50000
#define NT        (N_USERS + N_ITEMS)
#define DIM       64
#define INTENT_K  128
#define NEDGE     1500000
#define BATCH     4096
#define N_LAYERS  3
#define REG_LAMBDA 0.0001f
#define SSL_LAMBDA 0.2f
#define INV_TEMP   5.0f   // 1 / 0.2

// ---------------------------------------------------------------- utilities
__global__ void k_zero(float* __restrict__ p, int n) {
    int i = blockIdx.x * blockDim.x + threadIdx.x;
    if (i < n) p[i] = 0.0f;
}

__global__ void k_concat(const float* __restrict__ ue, const float* __restrict__ ie,
                         float* __restrict__ x) {
    int i = blockIdx.x * blockDim.x + threadIdx.x;
    if (i < NT * DIM) x[i] = (i < N_USERS * DIM) ? ue[i] : ie[i - N_USERS * DIM];
}

__global__ void k_add(float* __restrict__ g, const float* __restrict__ x, int n) {
    int i = blockIdx.x * blockDim.x + threadIdx.x;
    if (i < n) g[i] += x[i];
}

// ---------------------------------------------------------------- graph prep
__global__ void k_deg(const int* __restrict__ eu, const int* __restrict__ ei,
                      float* __restrict__ du, float* __restrict__ di) {
    int e = blockIdx.x * blockDim.x + threadIdx.x;
    if (e < NEDGE) {
        atomicAdd(&du[eu[e]], 1.0f);
        atomicAdd(&di[ei[e]], 1.0f);
    }
}

__global__ void k_vals(const int* __restrict__ eu, const int* __restrict__ ei,
                       const float* __restrict__ du, const float* __restrict__ di,
                       float* __restrict__ vals) {
    int e = blockIdx.x * blockDim.x + threadIdx.x;
    if (e < NEDGE) {
        float a = fmaxf(du[eu[e]], 1.0f);
        float b = fmaxf(di[ei[e]], 1.0f);
        vals[e] = 1.0f / sqrtf(a * b);
    }
}

// Symmetric normalized SpMM: 16 threads per edge, 4 floats per thread, both directions.
__global__ void k_spmm(const int* __restrict__ eu, const int* __restrict__ ei,
                       const float* __restrict__ vals,
                       const float* __restrict__ x, float* __restrict__ y) {
    long long idx = (long long)blockIdx.x * blockDim.x + threadIdx.x;
    if (idx >= (long long)NEDGE * 16) return;
    int e = (int)(idx >> 4);
    int c = ((int)idx & 15) * 4;
    int u = eu[e];
    int ir = N_USERS + ei[e];
    float v = vals[e];
    const float4 xu = *(const float4*)(x + (size_t)u * DIM + c);
    const float4 xi = *(const float4*)(x + (size_t)ir * DIM + c);
    float* yu = y + (size_t)u * DIM + c;
    float* yi = y + (size_t)ir * DIM + c;
    atomicAdd(yu + 0, v * xi.x); atomicAdd(yu + 1, v * xi.y);
    atomicAdd(yu + 2, v * xi.z); atomicAdd(yu + 3, v * xi.w);
    atomicAdd(yi + 0, v * xu.x); atomicAdd(yi + 1, v * xu.y);
    atomicAdd(yi + 2, v * xu.z); atomicAdd(yi + 3, v * xu.w);
}

// --------------------------------------------------- per-row intent + final
// One block (128 threads) per gathered row: intent = softmax(g@W) @ W^T,
// final = g + intent * noise.  W is row-major (DIM, INTENT_K).
__global__ void k_gather_final(const float* __restrict__ G, const float* __restrict__ W,
                               const float* __restrict__ noise, const int* __restrict__ idx,
                               int row_off, float* __restrict__ out_final,
                               float* __restrict__ out_intent) {
    __shared__ float sg[DIM];
    __shared__ float sp[INTENT_K];
    __shared__ float red[INTENT_K];
    int b = blockIdx.x;
    int t = threadIdx.x;               // 0..127
    int row = idx[b] + row_off;
    if (t < DIM) sg[t] = G[(size_t)row * DIM + t];
    __syncthreads();
    float logit = 0.0f;
    for (int d = 0; d < DIM; ++d) logit += sg[d] * W[d * INTENT_K + t];
    red[t] = logit; __syncthreads();
    for (int s = 64; s > 0; s >>= 1) { if (t < s) red[t] = fmaxf(red[t], red[t + s]); __syncthreads(); }
    float m = red[0]; __syncthreads();
    float ex = expf(logit - m);
    red[t] = ex; __syncthreads();
    for (int s = 64; s > 0; s >>= 1) { if (t < s) red[t] += red[t + s]; __syncthreads(); }
    float denom = red[0];
    sp[t] = ex / denom;
    __syncthreads();
    if (t < DIM) {
        float it = 0.0f;
        for (int k = 0; k < INTENT_K; ++k) it += sp[k] * W[t * INTENT_K + k];
        float fin = sg[t] + it * noise[(size_t)row * DIM + t];
        out_final[(size_t)b * DIM + t] = fin;
        if (out_intent) out_intent[(size_t)b * DIM + t] = it;
    }
}

// One wave per row: xn = x / ||x||
__global__ void k_normalize(const float* __restrict__ x, float* __restrict__ xn) {
    int b = blockIdx.x;
    int lane = threadIdx.x;            // 32 threads
    float a0 = x[(size_t)b * DIM + lane];
    float a1 = x[(size_t)b * DIM + lane + 32];
    float s = a0 * a0 + a1 * a1;
    for (int off = 16; off > 0; off >>= 1) s += __shfl_xor(s, off);
    float inv = 1.0f / sqrtf(s);
    xn[(size_t)b * DIM + lane]      = a0 * inv;
    xn[(size_t)b * DIM + lane + 32] = a1 * inv;
}

// ---------------------------------------------------------------- BPR + reg
__global__ void k_bpr(const float* __restrict__ ue, const float* __restrict__ pe,
                      const float* __restrict__ ne, float* __restrict__ out) {
    int b = blockIdx.x * blockDim.x + threadIdx.x;  // 4096 threads
    float ps = 0.0f, ns = 0.0f;
    for (int d = 0; d < DIM; ++d) {
        float u = ue[(size_t)b * DIM + d];
        ps += u * pe[(size_t)b * DIM + d];
        ns += u * ne[(size_t)b * DIM + d];
    }
    float x = ns - ps;
    float sp = fmaxf(x, 0.0f) + log1pf(expf(-fabsf(x)));
    __shared__ float red[256];
    red[threadIdx.x] = sp; __syncthreads();
    for (int s = 128; s > 0; s >>= 1) { if (threadIdx.x < s) red[threadIdx.x] += red[threadIdx.x + s]; __syncthreads(); }
    if (threadIdx.x == 0) atomicAdd(out, red[0] * (1.0f / BATCH));
}

#define NREG (3 * BATCH * DIM + 2 * DIM * INTENT_K)
__global__ void k_reg(const float* __restrict__ uemb, const float* __restrict__ iemb,
                      const float* __restrict__ uw, const float* __restrict__ iw,
                      const int* __restrict__ user, const int* __restrict__ pos,
                      const int* __restrict__ neg, float* __restrict__ out) {
    int i = blockIdx.x * blockDim.x + threadIdx.x;
    float v = 0.0f;
    if (i < 3 * BATCH * DIM) {
        int which = i / (BATCH * DIM);
        int r = (i / DIM) % BATCH;
        int d = i % DIM;
        if (which == 0)      v = uemb[(size_t)user[r] * DIM + d];
        else if (which == 1) v = iemb[(size_t)pos[r] * DIM + d];
        else                 v = iemb[(size_t)neg[r] * DIM + d];
    } else if (i < NREG) {
        int j = i - 3 * BATCH * DIM;
        v = (j < DIM * INTENT_K) ? uw[j] : iw[j - DIM * INTENT_K];
    }
    v = v * v;
    __shared__ float red[256];
    red[threadIdx.x] = v; __syncthreads();
    for (int s = 128; s > 0; s >>= 1) { if (threadIdx.x < s) red[threadIdx.x] += red[threadIdx.x + s]; __syncthreads(); }
    if (threadIdx.x == 0) atomicAdd(out, red[0] * (0.5f * REG_LAMBDA / BATCH));
}

// ------------------------------------------------- InfoNCE logits via WMMA
// Block = 256 threads (8 waves); block owns 16 rows; wave w handles column
// tiles w, w+8, ...  Computes expsum[row] += sum_j exp(5*dot - 5).
__global__ __launch_bounds__(256) void k_infonce_gemm(const float* __restrict__ Xn,
                                                      const float* __restrict__ Yn,
                                                      float* __restrict__ expsum) {
    int wave = threadIdx.x >> 5;
    int lane = threadIdx.x & 31;
    int m    = lane & 15;
    int half = lane >> 4;
    int rowBase = blockIdx.x * 16;

    // A fragment: lane m (half 0) holds A[m][k],A[m][k+1]; half 1 holds k+2,k+3.
    v2f a[16];
    const float* arow = Xn + (size_t)(rowBase + m) * DIM + half * 2;
#pragma unroll
    for (int kt = 0; kt < 16; ++kt) a[kt] = *(const v2f*)(arow + kt * 4);

    float acc[8];
#pragma unroll
    for (int r = 0; r < 8; ++r) acc[r] = 0.0f;

    for (int ct = wave; ct < BATCH / 16; ct += 8) {
        int colBase = ct * 16;
        const float* brow = Yn + (size_t)(colBase + m) * DIM + half * 2;
        v8f c = 0.0f;
#pragma unroll
        for (int kt = 0; kt < 16; ++kt) {
            v2f bf = *(const v2f*)(brow + kt * 4);
            c = __builtin_amdgcn_wmma_f32_16x16x4_f32(false, a[kt], false, bf,
                                                      (short)0, c, false, false);
        }
#pragma unroll
        for (int r = 0; r < 8; ++r)
            acc[r] += __expf(c[r] * INV_TEMP - 5.0f);   // shift: logits <= 5
    }

    // reduce over the 16 lanes of each half (each lane owns one column n)
#pragma unroll
    for (int r = 0; r < 8; ++r) {
        float v = acc[r];
        v += __shfl_xor(v, 1); v += __shfl_xor(v, 2);
        v += __shfl_xor(v, 4); v += __shfl_xor(v, 8);
        acc[r] = v;
    }
    if ((lane & 15) == 0) {
#pragma unroll
        for (int r = 0; r < 8; ++r)
            atomicAdd(&expsum[rowBase + r + half * 8], acc[r]);
    }
}

// Per row: lse - pos = log(expsum) + 5 - 5*dot(xn_i, yn_i); mean * SSL_LAMBDA.
__global__ void k_infonce_fin(const float* __restrict__ Xn, const float* __restrict__ Yn,
                              const float* __restrict__ expsum, float* __restrict__ out) {
    int i = blockIdx.x * blockDim.x + threadIdx.x;  // 4096 threads
    float v = 0.0f;
    if (i < BATCH) {
        float dot = 0.0f;
        for (int d = 0; d < DIM; ++d) dot += Xn[(size_t)i * DIM + d] * Yn[(size_t)i * DIM + d];
        v = logf(expsum[i]) + 5.0f - dot * INV_TEMP;
    }
    __shared__ float red[256];
    red[threadIdx.x] = v; __syncthreads();
    for (int s = 128; s > 0; s >>= 1) { if (threadIdx.x < s) red[threadIdx.x] += red[threadIdx.x + s]; __syncthreads(); }
    if (threadIdx.x == 0) atomicAdd(out, red[0] * (SSL_LAMBDA / BATCH));
}

// ---------------------------------------------------------------- launcher
static inline int nblk(long long n, int t) { return (int)((n + t - 1) / t); }

extern "C" void kernel_launch(void* const* d_in, const int* in_sizes, int n_in,
                              void* d_out, int out_size, void* d_ws, size_t ws_size,
                              hipStream_t stream) {
    const float* user_emb = (const float*)d_in[0];
    const float* item_emb = (const float*)d_in[1];
    const float* uw       = (const float*)d_in[2];
    const float* iw       = (const float*)d_in[3];
    const float* noise    = (const float*)d_in[4];
    const int*   eu       = (const int*)d_in[5];
    const int*   ei       = (const int*)d_in[6];
    const int*   user     = (const int*)d_in[7];
    const int*   pos      = (const int*)d_in[8];
    const int*   neg      = (const int*)d_in[9];
    float* out = (float*)d_out;

    float* ws = (float*)d_ws;
    size_t o = 0;
    float* xA   = ws + o; o += (size_t)NT * DIM;
    float* xB   = ws + o; o += (size_t)NT * DIM;
    float* G    = ws + o; o += (size_t)NT * DIM;
    float* degu = ws + o; o += N_USERS;
    float* degi = ws + o; o += N_ITEMS;
    float* vals = ws + o; o += NEDGE;
    float* ue   = ws + o; o += (size_t)BATCH * DIM;
    float* pe   = ws + o; o += (size_t)BATCH * DIM;
    float* ne   = ws + o; o += (size_t)BATCH * DIM;
    float* iu   = ws + o; o += (size_t)BATCH * DIM;
    float* ip   = ws + o; o += (size_t)BATCH * DIM;
    float* uen  = ws + o; o += (size_t)BATCH * DIM;
    float* pen  = ws + o; o += (size_t)BATCH * DIM;
    float* iun  = ws + o; o += (size_t)BATCH * DIM;
    float* ipn  = ws + o; o += (size_t)BATCH * DIM;
    float* es   = ws + o; o += 5 * BATCH;

    const int T = 256;
    // zero accumulators
    k_zero<<<1, 32, 0, stream>>>(out, 3);
    k_zero<<<nblk(NT, T), T, 0, stream>>>(degu, NT);            // degu+degi contiguous
    k_zero<<<nblk(5 * BATCH, T), T, 0, stream>>>(es, 5 * BATCH);
    k_zero<<<nblk((long long)NT * DIM, T), T, 0, stream>>>(G, NT * DIM);

    // graph prep
    k_concat<<<nblk((long long)NT * DIM, T), T, 0, stream>>>(user_emb, item_emb, xA);
    k_deg<<<nblk(NEDGE, T), T, 0, stream>>>(eu, ei, degu, degi);
    k_vals<<<nblk(NEDGE, T), T, 0, stream>>>(eu, ei, degu, degi, vals);

    // 3-layer propagation, gnn accumulation
    float* cur = xA; float* nxt = xB;
    for (int l = 0; l < N_LAYERS; ++l) {
        k_zero<<<nblk((long long)NT * DIM, T), T, 0, stream>>>(nxt, NT * DIM);
        k_spmm<<<nblk((long long)NEDGE * 16, T), T, 0, stream>>>(eu, ei, vals, cur, nxt);
        k_add<<<nblk((long long)NT * DIM, T), T, 0, stream>>>(G, nxt, NT * DIM);
        float* t2 = cur; cur = nxt; nxt = t2;
    }

    // gathered intent + final rows
    k_gather_final<<<BATCH, 128, 0, stream>>>(G, uw, noise, user, 0,       ue, iu);
    k_gather_final<<<BATCH, 128, 0, stream>>>(G, iw, noise, pos, N_USERS,  pe, ip);
    k_gather_final<<<BATCH, 128, 0, stream>>>(G, iw, noise, neg, N_USERS,  ne, nullptr);

    // normalized copies for the SSL terms
    k_normalize<<<BATCH, 32, 0, stream>>>(ue, uen);
    k_normalize<<<BATCH, 32, 0, stream>>>(pe, pen);
    k_normalize<<<BATCH, 32, 0, stream>>>(iu, iun);
    k_normalize<<<BATCH, 32, 0, stream>>>(ip, ipn);

    // bpr + reg
    k_bpr<<<BATCH / 256, 256, 0, stream>>>(ue, pe, ne, out);
    k_reg<<<nblk(NREG, T), T, 0, stream>>>(user_emb, item_emb, uw, iw, user, pos, neg, out + 1);

    // 5 InfoNCE terms
    const float* tx[5] = {uen, pen, uen, iun, ipn};
    const float* ty[5] = {uen, pen, pen, iun, ipn};
    for (int t = 0; t < 5; ++t) {
        float* est = es + t * BATCH;
        k_infonce_gemm<<<BATCH / 16, 256, 0, stream>>>(tx[t], ty[t], est);
        k_infonce_fin<<<BATCH / 256, 256, 0, stream>>>(tx[t], ty[t], est, out + 2);
    }
}